// GCNClassifier_3315714752625
// MI455X (gfx1250) — compile-verified
//
#include <hip/hip_runtime.h>
#include <hip/hip_bf16.h>
#include <math.h>

typedef __attribute__((ext_vector_type(2))) float v2f;
typedef __attribute__((ext_vector_type(8))) float v8f;

#define NN 50000
#define EE 800000
#define GG 128
#define INC 128
#define HH 128
#define CC 10

// ---------------- utility ----------------
__global__ void k_zero(float* __restrict__ p, long n) {
  long i = (long)blockIdx.x * blockDim.x + threadIdx.x;
  if (i < n) p[i] = 0.0f;
}

// deg = segment_sum(ew, dst) over real edges + 1.0 self-loop per node
__global__ void k_degree(const int* __restrict__ src, const int* __restrict__ dst,
                         const float* __restrict__ ew, float* __restrict__ deg,
                         int nE, int nN) {
  int i = blockIdx.x * blockDim.x + threadIdx.x;
  if (i < nE) {
    atomicAdd(&deg[dst[i]], ew[i]);
  } else if (i < nE + nN) {
    atomicAdd(&deg[i - nE], 1.0f);
  }
}

__global__ void k_dinv(const float* __restrict__ deg, float* __restrict__ dinv, int nN) {
  int i = blockIdx.x * blockDim.x + threadIdx.x;
  if (i < nN) {
    float d = deg[i];
    dinv[i] = (d > 0.0f) ? rsqrtf(d) : 0.0f;
  }
}

// zero-pad W3 (128 x 10) into Wp (128 x 16) so GEMM B-loads are unconditional
__global__ void k_padW3(const float* __restrict__ W3, float* __restrict__ Wp) {
  int i = blockIdx.x * blockDim.x + threadIdx.x;
  if (i < 128 * 16) {
    int r = i >> 4, c = i & 15;
    Wp[i] = (c < CC) ? W3[r * CC + c] : 0.0f;
  }
}

// ---------------- fp32 WMMA GEMM: Out[M x NOUT] = X[M x 128] @ W[128 x NCOLS] ----------------
// NCOLS/NOUT compile-time: no runtime masks, 32-bit shift addressing.
// Block = 256 threads (8 waves); block stages its 16 x 128 A tile in LDS (padded stride 132
// -> conflict-free ds_load_b64 A-fragments), each wave owns one 16x16 output tile.
// M = 50000 = 3125 * 16 exactly; EXEC is all-ones at every WMMA.
template <int NCOLS, int NOUT>
__global__ void __launch_bounds__(256)
k_gemm_wmma(const float* __restrict__ X, const float* __restrict__ W,
            float* __restrict__ Out) {
  constexpr int LDA = 132;               // padded LDS row stride (floats)
  __shared__ float As[16 * LDA];

  const int t    = threadIdx.x;
  const int row0 = blockIdx.x * 16;

  // ---- cooperative stage: 16 rows x 128 floats, 8 floats per thread ----
  {
    const int idx = t * 8;               // 0..2047
    const int r = idx >> 7;              // / 128
    const int c = idx & 127;
    const float4* g = (const float4*)(X + (row0 + r) * 128 + c);
    float4 v0 = g[0];
    float4 v1 = g[1];
    float* l = &As[r * LDA + c];
    *(float4*)(l + 0) = v0;
    *(float4*)(l + 4) = v1;
  }
  __syncthreads();

  const int wave = t >> 5;
  const int col0 = wave * 16;
  if (col0 < NCOLS) {                    // wave-uniform
    const int lane = t & 31;
    const int half = lane >> 4;          // 0: K pair {0,1}, 1: K pair {2,3}
    const int lm   = lane & 15;          // m for A-frag, n for B-frag
    const int n    = col0 + lm;

    const float* __restrict__ wcol = W + n;       // stride NCOLS between K rows
    const float* arow = &As[lm * LDA];

    v8f acc = {};
#pragma unroll
    for (int k0 = 0; k0 < 128; k0 += 4) {
      const int ka = k0 + half * 2;
      v2f a = *(const v2f*)(arow + ka);           // ds_load_b64
      v2f b;
      b.x = wcol[ka * NCOLS];
      b.y = wcol[(ka + 1) * NCOLS];
      // (neg_a, A, neg_b, B, c_mod, C, reuse_a, reuse_b)
      acc = __builtin_amdgcn_wmma_f32_16x16x4_f32(false, a, false, b, (short)0, acc,
                                                  false, false);
    }
    if (NOUT == NCOLS || n < NOUT) {
#pragma unroll
      for (int r = 0; r < 8; ++r) {
        const int row = row0 + half * 8 + r;      // VGPR r: M=r (lanes 0-15), M=r+8 (16-31)
        Out[row * NOUT + n] = acc[r];
      }
    }
  }
}

// ---------------- aggregation: agg[dst] += h[src] * (dinv[s]*w*dinv[d]), 128 feats ----------------
// One wave32 per (edge or self-loop); float4 gather per lane (512B coalesced row), f32 atomics scatter.
__global__ void __launch_bounds__(256)
k_agg128(const float* __restrict__ h, const int* __restrict__ src, const int* __restrict__ dst,
         const float* __restrict__ ew, const float* __restrict__ dinv,
         float* __restrict__ agg, int nE, int nN) {
  const long gt  = (long)blockIdx.x * blockDim.x + threadIdx.x;
  const int  wid = (int)(gt >> 5);
  const int lane = threadIdx.x & 31;
  if (wid >= nE + nN) return;
  int s, d; float w;
  if (wid < nE) { s = src[wid]; d = dst[wid]; w = ew[wid]; }
  else          { s = wid - nE; d = s;        w = 1.0f;    }
  const float nrm = dinv[s] * w * dinv[d];
  const float4 v = ((const float4*)(h + (size_t)s * HH))[lane];
  float* ad = agg + (size_t)d * HH + lane * 4;
  atomicAdd(ad + 0, v.x * nrm);
  atomicAdd(ad + 1, v.y * nrm);
  atomicAdd(ad + 2, v.z * nrm);
  atomicAdd(ad + 3, v.w * nrm);
}

// 10-feature variant for the final layer: one thread per edge
__global__ void k_aggC(const float* __restrict__ h, const int* __restrict__ src,
                       const int* __restrict__ dst, const float* __restrict__ ew,
                       const float* __restrict__ dinv, float* __restrict__ agg,
                       int nE, int nN) {
  const int e = blockIdx.x * blockDim.x + threadIdx.x;
  if (e >= nE + nN) return;
  int s, d; float w;
  if (e < nE) { s = src[e]; d = dst[e]; w = ew[e]; }
  else        { s = e - nE; d = s;      w = 1.0f;  }
  const float nrm = dinv[s] * w * dinv[d];
#pragma unroll
  for (int c = 0; c < CC; ++c)
    atomicAdd(&agg[(size_t)d * CC + c], h[(size_t)s * CC + c] * nrm);
}

// t[i] = (t[i] + bias[i % feat]) with optional ReLU
__global__ void k_bias_act(float* __restrict__ t, const float* __restrict__ b,
                           long n, int feat, int do_relu) {
  long i = (long)blockIdx.x * blockDim.x + threadIdx.x;
  if (i >= n) return;
  float v = t[i] + b[(int)(i % feat)];
  t[i] = do_relu ? fmaxf(v, 0.0f) : v;
}

// global mean pool numerators + counts
__global__ void k_pool(const float* __restrict__ h3, const int* __restrict__ batch,
                       float* __restrict__ pooled, float* __restrict__ cnt, int nN) {
  int i = blockIdx.x * blockDim.x + threadIdx.x;
  if (i >= nN) return;
  const int g = batch[i];
  atomicAdd(&cnt[g], 1.0f);
#pragma unroll
  for (int c = 0; c < CC; ++c)
    atomicAdd(&pooled[(size_t)g * CC + c], h3[(size_t)i * CC + c]);
}

__global__ void k_logsoftmax(const float* __restrict__ pooled, const float* __restrict__ cnt,
                             float* __restrict__ out, int nG) {
  int g = blockIdx.x * blockDim.x + threadIdx.x;
  if (g >= nG) return;
  const float c = fmaxf(cnt[g], 1.0f);
  float v[CC];
  float mx = -INFINITY;
#pragma unroll
  for (int i = 0; i < CC; ++i) { v[i] = pooled[(size_t)g * CC + i] / c; mx = fmaxf(mx, v[i]); }
  float sum = 0.0f;
#pragma unroll
  for (int i = 0; i < CC; ++i) sum += expf(v[i] - mx);
  const float lse = mx + logf(sum);
#pragma unroll
  for (int i = 0; i < CC; ++i) out[(size_t)g * CC + i] = v[i] - lse;
}

extern "C" void kernel_launch(void* const* d_in, const int* in_sizes, int n_in,
                              void* d_out, int out_size, void* d_ws, size_t ws_size,
                              hipStream_t stream) {
  const float* x     = (const float*)d_in[0];
  const int*   ei    = (const int*)d_in[1];
  const float* ea    = (const float*)d_in[2];
  const int*   batch = (const int*)d_in[3];
  const float* W1 = (const float*)d_in[4]; const float* b1 = (const float*)d_in[5];
  const float* W2 = (const float*)d_in[6]; const float* b2 = (const float*)d_in[7];
  const float* W3 = (const float*)d_in[8]; const float* b3 = (const float*)d_in[9];
  const int* src = ei;        // edge_index[0]
  const int* dst = ei + EE;   // edge_index[1]
  float* out = (float*)d_out;

  // workspace layout (floats); deg+dinv+Wpad = 102048 floats keeps t0/t1 16B-aligned
  float* ws = (float*)d_ws;
  size_t off = 0;
  float* deg    = ws + off; off += NN;
  float* dinv   = ws + off; off += NN;
  float* Wpad   = ws + off; off += 128 * 16;          // zero-padded W3
  float* t0     = ws + off; off += (size_t)NN * HH;   // GEMM output
  float* t1     = ws + off; off += (size_t)NN * HH;   // aggregation output / next input
  float* pooled = ws + off; off += (size_t)GG * CC;
  float* cnt    = ws + off; off += GG;

  const dim3 B(256);
  auto nb = [](long n) { return dim3((unsigned)((n + 255) / 256)); };
  const int Mtiles = NN / 16;  // 3125, exact

  // normalization terms + padded W3
  k_zero<<<nb(NN), B, 0, stream>>>(deg, NN);
  k_degree<<<nb(EE + NN), B, 0, stream>>>(src, dst, ea, deg, EE, NN);
  k_dinv<<<nb(NN), B, 0, stream>>>(deg, dinv, NN);
  k_padW3<<<nb(128 * 16), B, 0, stream>>>(W3, Wpad);

  // layer 1: relu(agg(x@W1) + b1)
  k_gemm_wmma<HH, HH><<<dim3(Mtiles), B, 0, stream>>>(x, W1, t0);
  k_zero<<<nb((long)NN * HH), B, 0, stream>>>(t1, (long)NN * HH);
  k_agg128<<<nb((long)(EE + NN) * 32), B, 0, stream>>>(t0, src, dst, ea, dinv, t1, EE, NN);
  k_bias_act<<<nb((long)NN * HH), B, 0, stream>>>(t1, b1, (long)NN * HH, HH, 1);

  // layer 2: relu(agg(h1@W2) + b2)
  k_gemm_wmma<HH, HH><<<dim3(Mtiles), B, 0, stream>>>(t1, W2, t0);
  k_zero<<<nb((long)NN * HH), B, 0, stream>>>(t1, (long)NN * HH);
  k_agg128<<<nb((long)(EE + NN) * 32), B, 0, stream>>>(t0, src, dst, ea, dinv, t1, EE, NN);
  k_bias_act<<<nb((long)NN * HH), B, 0, stream>>>(t1, b2, (long)NN * HH, HH, 1);

  // layer 3: agg(h2@W3) + b3   (C = 10, W zero-padded to 16 cols)
  k_gemm_wmma<16, CC><<<dim3(Mtiles), B, 0, stream>>>(t1, Wpad, t0);
  k_zero<<<nb((long)NN * CC), B, 0, stream>>>(t1, (long)NN * CC);
  k_aggC<<<nb(EE + NN), B, 0, stream>>>(t0, src, dst, ea, dinv, t1, EE, NN);
  k_bias_act<<<nb((long)NN * CC), B, 0, stream>>>(t1, b3, (long)NN * CC, CC, 0);

  // global mean pool + log_softmax
  k_zero<<<nb((long)GG * CC + GG), B, 0, stream>>>(pooled, (long)GG * CC + GG); // pooled & cnt contiguous
  k_pool<<<nb(NN), B, 0, stream>>>(t1, batch, pooled, cnt, NN);
  k_logsoftmax<<<nb(GG), B, 0, stream>>>(pooled, cnt, out, GG);
}